// get_reward_4758823764602
// MI455X (gfx1250) — compile-verified
//
#include <hip/hip_runtime.h>
#include <math.h>

// ---------------------------------------------------------------------------
// Pairwise-distance + min/max loss via V_WMMA_F32_16X16X4_F32 (CDNA5 wave32).
//
//   ||t - s||^2 = ||t||^2 - 2 t.s + ||s||^2
//
// WMMA tile: M = 16 targets (rows), N = 16 sources (cols), K = 4 (xyz + 0 pad).
//   A (16x4, 2 VGPRs): lane L<16 -> (-2x,-2y) of target row L
//                      lane L>=16 -> (-2z, 0 ) of target row L-16   (K=2,3)
//   B (4x16, 2 VGPRs): lane L, col n=L%16: L<16 -> (x,y) of source n
//                                          L>=16 -> (z,0) of source n
//   C/D (16x16 f32, 8 VGPRs, documented): lane L = col n=L%16,
//                      VGPR r = row m = 8*(L/16)+r.
// C is preloaded with ||t_m||^2, so D = ||t||^2 - 2 t.s.  ||s||^2 is added
// after the min/max reduction (constant per column n).
// ---------------------------------------------------------------------------

typedef float v2f __attribute__((ext_vector_type(2)));
typedef float v4f __attribute__((ext_vector_type(4)));
typedef float v8f __attribute__((ext_vector_type(8)));

#define CHUNK_TILES 128               // 2048 targets per LDS chunk
#define K1_THREADS 256                // 8 waves -> 128 source points / block

__global__ __launch_bounds__(K1_THREADS)
void loss2_wmma_kernel(const float* __restrict__ source,
                       const float* __restrict__ target,
                       const float* __restrict__ trans_end,
                       float* __restrict__ loss_start,
                       float* __restrict__ loss_end,
                       int N, int M)
{
    // A-layout targets (pre-scaled by -2) and ||t||^2 per chunk.
    __shared__ __align__(16) float sA[CHUNK_TILES * 32 * 2];   // 32 KB
    __shared__ __align__(32) float sTn[CHUNK_TILES * 16];      //  8 KB

    const int tid  = threadIdx.x;
    const int lane = tid & 31;
    const int wave = tid >> 5;
    const int col  = lane & 15;
    const int hi   = lane >> 4;                 // 0: lanes 0-15, 1: lanes 16-31
    const int n    = blockIdx.x * (K1_THREADS / 2) + wave * 16 + col;

    // Per-lane source point (start + denoised).
    const float sx = source[3 * n + 0];
    const float sy = source[3 * n + 1];
    const float sz = source[3 * n + 2];
    const float ex = sx - trans_end[3 * n + 0];
    const float ey = sy - trans_end[3 * n + 1];
    const float ez = sz - trans_end[3 * n + 2];

    v2f bS, bE;
    bS[0] = hi ? sz : sx;  bS[1] = hi ? 0.0f : sy;
    bE[0] = hi ? ez : ex;  bE[1] = hi ? 0.0f : ey;

    const float BIG = 3.402823466e+38f;
    float minS = BIG, maxS = -BIG, minE = BIG, maxE = -BIG;

    for (int base = 0; base < M; base += CHUNK_TILES * 16) {
        __syncthreads();
        // Cooperative fill: 2048 targets / 256 threads = 8 per thread.
        for (int k = tid; k < CHUNK_TILES * 16; k += K1_THREADS) {
            const int gi = base + k;
            const float tx = target[3 * gi + 0];
            const float ty = target[3 * gi + 1];
            const float tz = target[3 * gi + 2];
            const int t = k >> 4, c = k & 15;
            sA[(t * 32 + c) * 2 + 0]        = -2.0f * tx;
            sA[(t * 32 + c) * 2 + 1]        = -2.0f * ty;
            sA[(t * 32 + c + 16) * 2 + 0]   = -2.0f * tz;
            sA[(t * 32 + c + 16) * 2 + 1]   = 0.0f;
            sTn[t * 16 + c] = tx * tx + ty * ty + tz * tz;
        }
        __syncthreads();

        #pragma unroll 4
        for (int t = 0; t < CHUNK_TILES; ++t) {
            const v2f a = *(const v2f*)&sA[(t * 32 + lane) * 2];      // ds_load_b64
            const v8f c = *(const v8f*)&sTn[t * 16 + hi * 8];         // 2x ds_load_b128

            // D = A*B + C : tnorm - 2 t.s  (two passes share A and C)
            v8f dS = __builtin_amdgcn_wmma_f32_16x16x4_f32(
                false, a, false, bS, (short)0, c, false, false);
            v8f dE = __builtin_amdgcn_wmma_f32_16x16x4_f32(
                false, a, false, bE, (short)0, c, false, false);

            #pragma unroll
            for (int r = 0; r < 8; ++r) {
                minS = fminf(minS, dS[r]);  maxS = fmaxf(maxS, dS[r]);
                minE = fminf(minE, dE[r]);  maxE = fmaxf(maxE, dE[r]);
            }
        }
    }

    // Combine row-halves (rows 0-7 in lanes 0-15 vs rows 8-15 in lanes 16-31).
    minS = fminf(minS, __shfl_xor(minS, 16, 32));
    maxS = fmaxf(maxS, __shfl_xor(maxS, 16, 32));
    minE = fminf(minE, __shfl_xor(minE, 16, 32));
    maxE = fmaxf(maxE, __shfl_xor(maxE, 16, 32));

    if (hi == 0) {
        loss_start[n] = 0.99f * minS + 0.01f * maxS + (sx * sx + sy * sy + sz * sz);
        loss_end[n]   = 0.99f * minE + 0.01f * maxE + (ex * ex + ey * ey + ez * ez);
    }
}

// ---------------------------------------------------------------------------
// Per-point discounted-return scan (T steps, backwards) + mean reduction.
// ---------------------------------------------------------------------------
__global__ __launch_bounds__(256)
void reward_scan_kernel(const float* __restrict__ loss_start,
                        const float* __restrict__ loss_end,
                        const float* __restrict__ label,
                        const float* __restrict__ path_m,
                        const float* __restrict__ path_maxprob_m,
                        float* __restrict__ out,
                        int N, int T)
{
    __shared__ float red[256];
    const int i = blockIdx.x * blockDim.x + threadIdx.x;

    float val = 0.0f;
    if (i < N) {
        const float p = 0.002f, L0 = 0.4f, lammda = 0.05f, gamma = 0.99f;
        const float ls = loss_start[i];
        const float le = loss_end[i];
        const float loss_dt = le - ls;
        const float d = fminf(le / L0, 1.0f);
        const float l = __expf(-label[i]);
        const float coef = d + lammda * l;

        float R = 0.0f, Lc = 0.0f;
        for (int t = T - 1; t >= 0; --t) {
            float r = -p * path_m[t * N + i];
            if (t == T - 1) r -= coef * loss_dt;
            const float lp = __logf(path_maxprob_m[t * N + i]);
            R  = gamma * R + lp * (r + 0.02f);
            Lc = Lc - R;
        }
        val = Lc / (float)T / (float)N;      // contributes to the mean
    }

    red[threadIdx.x] = val;
    __syncthreads();
    for (int s = 128; s > 0; s >>= 1) {
        if (threadIdx.x < s) red[threadIdx.x] += red[threadIdx.x + s];
        __syncthreads();
    }
    if (threadIdx.x == 0) atomicAdd(out, red[0]);
}

extern "C" void kernel_launch(void* const* d_in, const int* in_sizes, int n_in,
                              void* d_out, int out_size, void* d_ws, size_t ws_size,
                              hipStream_t stream)
{
    const float* source         = (const float*)d_in[0];   // [N,3]
    const float* target         = (const float*)d_in[1];   // [M,3]
    const float* label          = (const float*)d_in[2];   // [N]
    const float* trans_m        = (const float*)d_in[3];   // [T,N,3]
    const float* path_m         = (const float*)d_in[4];   // [T,N]
    const float* path_maxprob_m = (const float*)d_in[5];   // [T,N]

    const int N = in_sizes[2];
    const int M = in_sizes[1] / 3;
    const int T = in_sizes[4] / N;

    const float* trans_end = trans_m + (size_t)(T - 1) * N * 3;

    float* loss_start = (float*)d_ws;        // [N]
    float* loss_end   = loss_start + N;      // [N]
    float* out        = (float*)d_out;

    hipMemsetAsync(out, 0, sizeof(float), stream);

    loss2_wmma_kernel<<<dim3(N / (K1_THREADS / 2)), dim3(K1_THREADS), 0, stream>>>(
        source, target, trans_end, loss_start, loss_end, N, M);

    reward_scan_kernel<<<dim3((N + 255) / 256), dim3(256), 0, stream>>>(
        loss_start, loss_end, label, path_m, path_maxprob_m, out, N, T);
}